// MGCUNET_LSTM_MIL_fusion2_86285892976682
// MI455X (gfx1250) — compile-verified
//
#include <hip/hip_runtime.h>
#include <math.h>

// Problem constants (from reference)
#define Bn   8
#define Tn   64
#define Hn   100      // LSTM hidden
#define Dn   64       // attention dim
#define Ln   100      // == H
#define NTn  512      // B*T
#define G4n  400      // 4*H
#define EPSV 1e-5f

typedef float v2f __attribute__((ext_vector_type(2)));
typedef float v4f __attribute__((ext_vector_type(4)));
typedef float v8f __attribute__((ext_vector_type(8)));

__device__ __forceinline__ float wave_sum(float v) {
#pragma unroll
  for (int o = 16; o > 0; o >>= 1) v += __shfl_xor(v, o, 32);
  return v;
}
__device__ __forceinline__ float sigm(float x) { return 1.0f / (1.0f + expf(-x)); }

// ---------------------------------------------------------------------------
// Kernel A: feat[b,t,r] = relu( dot(g[b,t,r,:], Wg) + bg )
// Bandwidth-bound: 1.13 GB streamed once (AI ~0.5 FLOP/B -> HBM roofline,
// ~48us floor at 23.3 TB/s). One wave per row; float4 non-temporal loads so
// each wave-instruction moves 512B (R is always a multiple of 4, rows stay
// 16B-aligned). Wg staged in LDS as float4.
// ---------------------------------------------------------------------------
__global__ void __launch_bounds__(256)
gcn_feat_kernel(const float* __restrict__ g, const float* __restrict__ Wg,
                const float* __restrict__ bg, float* __restrict__ feat,
                int R, int nRows) {
  __shared__ v4f sWg[128];                    // up to R=500 -> 125 float4
  const int R4 = R >> 2;
  for (int c = threadIdx.x; c < R4; c += blockDim.x)
    sWg[c] = ((const v4f*)Wg)[c];
  __syncthreads();
  const int wave = threadIdx.x >> 5;
  const int lane = threadIdx.x & 31;
  const int row  = blockIdx.x * (blockDim.x >> 5) + wave;
  if (row >= nRows) return;
  const v4f* rp = (const v4f*)(g + (size_t)row * R);
  float acc = 0.f;
#pragma unroll 2
  for (int c = lane; c < R4; c += 32) {
    const v4f gv = __builtin_nontemporal_load(rp + c);
    const v4f wv = sWg[c];
    acc += gv.x * wv.x + gv.y * wv.y + gv.z * wv.z + gv.w * wv.w;
  }
  acc = wave_sum(acc);
  if (lane == 0) feat[row] = fmaxf(acc + bg[0], 0.f);
}

// ---------------------------------------------------------------------------
// Kernel B: gates_in[n, j] = sum_c feat[n,c] * Wih[j,c] + bih[j] + bhh[j]
// Dense GEMM [512,R] x [R,400] done with V_WMMA_F32_16X16X4_F32.
// One wave per 16x16 output tile.  Lane layouts per CDNA5 ISA 7.12.2:
//   A 16x4 f32 : lane&15 = M, (lane>>4)*2 = K base, vgpr0/1 = K, K+1
//   B 4x16 f32 : lane&15 = N, (lane>>4)*2 = K base, vgpr0/1 = K, K+1
//   C 16x16 f32: vgpr v -> M = (lane>>4)*8 + v, N = lane&15
// ---------------------------------------------------------------------------
__global__ void __launch_bounds__(32)
lstm_ingemm_wmma(const float* __restrict__ X, const float* __restrict__ Wih,
                 const float* __restrict__ bih, const float* __restrict__ bhh,
                 float* __restrict__ Gout, int K) {
  const int lane  = threadIdx.x;
  const int mBase = blockIdx.x * 16;
  const int nBase = blockIdx.y * 16;
  const int mr = mBase + (lane & 15);
  const int nr = nBase + (lane & 15);
  const int k0 = (lane >> 4) << 1;              // 0 or 2
  const float* xp = X   + (size_t)mr * K + k0;  // A = feat (row-major, reduce over c)
  const float* wp = Wih + (size_t)nr * K + k0;  // B = Wih^T  (B[k][n] = Wih[n][k])
  v8f acc = {0.f, 0.f, 0.f, 0.f, 0.f, 0.f, 0.f, 0.f};
  for (int k = 0; k < K; k += 4) {
    v2f a; a.x = xp[k]; a.y = xp[k + 1];
    v2f b; b.x = wp[k]; b.y = wp[k + 1];
    acc = __builtin_amdgcn_wmma_f32_16x16x4_f32(false, a, false, b,
                                                (short)0, acc, false, false);
  }
  const int row = mBase + ((lane >> 4) << 3);
  const int col = nBase + (lane & 15);
  const float bias = bih[col] + bhh[col];
#pragma unroll
  for (int v = 0; v < 8; ++v)
    Gout[(size_t)(row + v) * G4n + col] = acc[v] + bias;
}

// ---------------------------------------------------------------------------
// Kernel C: per-branch LSTM recurrence + attention + batchnorm.
// One 1024-thread workgroup per branch. The full hidden-state history hs
// [512,100] (204.8 KB) stays in CDNA5's 320 KB WGP LDS -> no global traffic
// for the recurrent/attention phases. Whh streams from L2 (160 KB, hot).
// ---------------------------------------------------------------------------
struct BranchC {
  const float* Whh;   const float* Wa1; const float* ba1; const float* Wa2;
  const float* ba2;   const float* gamma; const float* beta;
  const float* gatesIn;
};
struct CArgs { BranchC br[5]; float* Mcat; };

__global__ void __launch_bounds__(1024)
lstm_attn_bn_kernel(CArgs args) {
  extern __shared__ float smem[];
  float* sHS = smem;                  // 512*100 = 51200 floats (hs history)
  float* sH  = sHS + NTn * Hn;        // 800 (current h)
  float* sC  = sH  + Bn * Hn;         // 800 (cell state)
  float* sG  = sC  + Bn * Hn;         // 3200 (gates for one step)
  float* sE  = sG  + Bn * G4n;        // 512 (attention scores / weights)
  float* sM  = sE  + NTn;             // 800 (pooled M)
  const BranchC& p = args.br[blockIdx.x];
  const int tid = threadIdx.x;

  for (int i = tid; i < Bn * Hn; i += blockDim.x) { sH[i] = 0.f; sC[i] = 0.f; }
  __syncthreads();

  for (int t = 0; t < Tn; ++t) {
    // gates[b][j] = gates_in[b*T+t][j] + Whh[j,:] . h_prev[b,:]
    for (int idx = tid; idx < Bn * G4n; idx += blockDim.x) {
      const int b = idx / G4n, j = idx - b * G4n;
      const float* wr = p.Whh + (size_t)j * Hn;
      const float* hp = sH + b * Hn;
      float acc = p.gatesIn[(size_t)(b * Tn + t) * G4n + j];
#pragma unroll 4
      for (int h = 0; h < Hn; ++h) acc += wr[h] * hp[h];
      sG[idx] = acc;
    }
    __syncthreads();
    // pointwise cell update (PyTorch gate order i,f,g,o)
    for (int idx = tid; idx < Bn * Hn; idx += blockDim.x) {
      const int b = idx / Hn, h = idx - b * Hn;
      const float gi = sigm(sG[b * G4n + h]);
      const float gf = sigm(sG[b * G4n + Hn + h]);
      const float gg = tanhf(sG[b * G4n + 2 * Hn + h]);
      const float go = sigm(sG[b * G4n + 3 * Hn + h]);
      const float c  = gf * sC[idx] + gi * gg;
      const float hn = go * tanhf(c);
      sC[idx] = c; sH[idx] = hn;
      sHS[(size_t)(b * Tn + t) * Hn + h] = hn;
    }
    __syncthreads();
  }

  // attention scores: e[b,t] = Wa2 . tanh(hs[b,t,:] @ Wa1 + ba1) + ba2
  const int wave = tid >> 5, lane = tid & 31, nw = blockDim.x >> 5;
  for (int pair = wave; pair < NTn; pair += nw) {
    const float* hv = sHS + (size_t)pair * Hn;
    float part = 0.f;
    for (int d = lane; d < Dn; d += 32) {
      float s = p.ba1[d];
      const float* wa = p.Wa1 + d;               // Wa1[l*Dn + d]
#pragma unroll 4
      for (int l = 0; l < Ln; ++l) s += hv[l] * wa[(size_t)l * Dn];
      part += p.Wa2[d] * tanhf(s);
    }
    part = wave_sum(part);
    if (lane == 0) sE[pair] = part + p.ba2[0];
  }
  __syncthreads();

  // softmax over T per batch row
  if (tid < Bn) {
    float mx = -1e30f;
    for (int t = 0; t < Tn; ++t) mx = fmaxf(mx, sE[tid * Tn + t]);
    float sum = 0.f;
    for (int t = 0; t < Tn; ++t) { float e = expf(sE[tid * Tn + t] - mx); sE[tid * Tn + t] = e; sum += e; }
    const float inv = 1.0f / sum;
    for (int t = 0; t < Tn; ++t) sE[tid * Tn + t] *= inv;
  }
  __syncthreads();

  // M[b,l] = sum_t alpha[b,t] * hs[b,t,l]
  for (int idx = tid; idx < Bn * Ln; idx += blockDim.x) {
    const int b = idx / Ln, l = idx - b * Ln;
    float m = 0.f;
    for (int t = 0; t < Tn; ++t) m += sE[b * Tn + t] * sHS[(size_t)(b * Tn + t) * Hn + l];
    sM[idx] = m;
  }
  __syncthreads();

  // BatchNorm over the batch dim (biased variance), write into concat buffer
  if (tid < Ln) {
    float mu = 0.f;
    for (int b = 0; b < Bn; ++b) mu += sM[b * Ln + tid];
    mu *= (1.0f / Bn);
    float var = 0.f;
    for (int b = 0; b < Bn; ++b) { const float d = sM[b * Ln + tid] - mu; var += d * d; }
    var *= (1.0f / Bn);
    const float sc = p.gamma[tid] * rsqrtf(var + EPSV);
    for (int b = 0; b < Bn; ++b)
      args.Mcat[(size_t)b * 500 + blockIdx.x * Ln + tid] = (sM[b * Ln + tid] - mu) * sc + p.beta[tid];
  }
}

// ---------------------------------------------------------------------------
// Kernel D: head MLP  [8,500] -> relu 256 -> relu 128 -> 2 -> softmax
// ---------------------------------------------------------------------------
__global__ void __launch_bounds__(256)
head_kernel(const float* __restrict__ Mcat,
            const float* __restrict__ Wf1, const float* __restrict__ bf1,
            const float* __restrict__ Wf2, const float* __restrict__ bf2,
            const float* __restrict__ Wc,  const float* __restrict__ bc,
            float* __restrict__ out) {
  __shared__ float sM[8 * 500];
  __shared__ float sH1[8 * 256];
  __shared__ float sH2[8 * 128];
  __shared__ float sL[16];
  const int tid = threadIdx.x;
  for (int i = tid; i < 8 * 500; i += 256) sM[i] = Mcat[i];
  __syncthreads();
  for (int idx = tid; idx < 8 * 256; idx += 256) {
    const int b = idx >> 8, j = idx & 255;
    float acc = bf1[j];
    for (int k = 0; k < 500; ++k) acc += sM[b * 500 + k] * Wf1[(size_t)k * 256 + j];
    sH1[idx] = fmaxf(acc, 0.f);
  }
  __syncthreads();
  for (int idx = tid; idx < 8 * 128; idx += 256) {
    const int b = idx >> 7, j = idx & 127;
    float acc = bf2[j];
    for (int k = 0; k < 256; ++k) acc += sH1[b * 256 + k] * Wf2[(size_t)k * 128 + j];
    sH2[idx] = fmaxf(acc, 0.f);
  }
  __syncthreads();
  if (tid < 16) {
    const int b = tid >> 1, j = tid & 1;
    float acc = bc[j];
    for (int k = 0; k < 128; ++k) acc += sH2[b * 128 + k] * Wc[k * 2 + j];
    sL[tid] = acc;
  }
  __syncthreads();
  if (tid < 8) {
    const float a = sL[tid * 2], c = sL[tid * 2 + 1];
    const float mx = fmaxf(a, c);
    const float ea = expf(a - mx), ec = expf(c - mx);
    const float inv = 1.0f / (ea + ec);
    out[tid * 2]     = ea * inv;
    out[tid * 2 + 1] = ec * inv;
  }
}

// ---------------------------------------------------------------------------
// Host launcher. Input order: g1..g5, then params flattened in setup_inputs()
// insertion order: branches[0..4] x {Wg,bg,Wih,Whh,bih,bhh,Wa1,ba1,Wa2,ba2,
// gamma,beta}, then Wf1,bf1,Wf2,bf2,Wc,bc.
// ---------------------------------------------------------------------------
extern "C" void kernel_launch(void* const* d_in, const int* in_sizes, int n_in,
                              void* d_out, int out_size, void* d_ws, size_t ws_size,
                              hipStream_t stream) {
  (void)in_sizes; (void)n_in; (void)out_size; (void)ws_size;
  const int Rs[5] = {100, 200, 300, 400, 500};
  const float* g[5];
  for (int i = 0; i < 5; ++i) g[i] = (const float*)d_in[i];
  auto bp = [&](int br, int k) -> const float* {
    return (const float*)d_in[5 + br * 12 + k];
  }; // 0:Wg 1:bg 2:Wih 3:Whh 4:bih 5:bhh 6:Wa1 7:ba1 8:Wa2 9:ba2 10:gamma 11:beta
  const float* Wf1 = (const float*)d_in[65];
  const float* bf1 = (const float*)d_in[66];
  const float* Wf2 = (const float*)d_in[67];
  const float* bf2 = (const float*)d_in[68];
  const float* Wc  = (const float*)d_in[69];
  const float* bc  = (const float*)d_in[70];

  float* ws = (float*)d_ws;
  size_t featOff[5], gatesOff[5], off = 0;
  for (int br = 0; br < 5; ++br) { featOff[br] = off; off += (size_t)NTn * Rs[br]; }
  for (int br = 0; br < 5; ++br) { gatesOff[br] = off; off += (size_t)NTn * G4n; }
  const size_t mcatOff = off;

  CArgs ca;
  ca.Mcat = ws + mcatOff;
  for (int br = 0; br < 5; ++br) {
    ca.br[br].Whh     = bp(br, 3);
    ca.br[br].Wa1     = bp(br, 6);
    ca.br[br].ba1     = bp(br, 7);
    ca.br[br].Wa2     = bp(br, 8);
    ca.br[br].ba2     = bp(br, 9);
    ca.br[br].gamma   = bp(br, 10);
    ca.br[br].beta    = bp(br, 11);
    ca.br[br].gatesIn = ws + gatesOff[br];
  }

  for (int br = 0; br < 5; ++br) {
    const int R = Rs[br];
    const int rows = NTn * R;
    gcn_feat_kernel<<<(rows + 7) / 8, 256, 0, stream>>>(
        g[br], bp(br, 0), bp(br, 1), ws + featOff[br], R, rows);
    dim3 gb(NTn / 16, G4n / 16);   // 32 x 25 tiles, one wave each
    lstm_ingemm_wmma<<<gb, 32, 0, stream>>>(
        ws + featOff[br], bp(br, 2), bp(br, 4), bp(br, 5), ws + gatesOff[br], R);
  }

  const size_t smemC =
      (size_t)(NTn * Hn + 2 * Bn * Hn + Bn * G4n + NTn + Bn * Ln) * sizeof(float);
  // Opt in to >64KB dynamic LDS (CDNA5 allows up to 320KB per workgroup).
  // Idempotent host-side config; safe to call on every launch (no stream op).
  hipFuncSetAttribute((const void*)lstm_attn_bn_kernel,
                      hipFuncAttributeMaxDynamicSharedMemorySize, (int)smemC);
  lstm_attn_bn_kernel<<<5, 1024, smemC, stream>>>(ca);

  head_kernel<<<1, 256, 0, stream>>>(ws + mcatOff, Wf1, bf1, Wf2, bf2, Wc, bc,
                                     (float*)d_out);
}